// HarmonicOscillator_83940840833424
// MI455X (gfx1250) — compile-verified
//
#include <hip/hip_runtime.h>

typedef __attribute__((ext_vector_type(2))) float v2f;
typedef __attribute__((ext_vector_type(8))) float v8f;

namespace {
constexpr int   kB  = 8;
constexpr int   kH  = 100;      // harmonics
constexpr int   kT  = 1000;     // frames
constexpr int   kTs = 80000;    // samples
constexpr int   kSamplesPerWG = 128;
constexpr int   kChunksPerB   = kTs / kSamplesPerWG;   // 625
// (T-1)/(Ts-1) rounded to fp32, matching the reference's fp32 index math
constexpr float kR = (float)(999.0 / 79999.0);
}

// signal[b,s] = sum_h sin( 2*pi*f0_up*s*(h+1)/sr + ps_up ) * hd_up[b,h,s]
//
// Per 16-sample wave tile:
//   P[frame, s] = sum_h hd[b,h,frame] * sin((h+1)*beta_s + phi_s)
// computed as a 16x16x100 fp32 GEMM via 25x V_WMMA_F32_16X16X4_F32,
// with the sine matrix generated by the Chebyshev recurrence
//   S_{h+1} = 2cos(beta)*S_h - S_{h-1}
// Then signal = (1-w)*P[m] + w*P[m+1].
__global__ __launch_bounds__(256) void harmosc_kernel(
    const float* __restrict__ f0, const float* __restrict__ hd,
    const float* __restrict__ ps, float* __restrict__ out) {
  const int blk   = blockIdx.x;
  const int b     = blk / kChunksPerB;
  const int chunk = blk - b * kChunksPerB;
  const int sBase = chunk * kSamplesPerWG;

  // workgroup-level base frame; i0(s) for s in this WG is in {t0, t0+1, t0+2}
  int t0 = (int)((float)sBase * kR);
  if (t0 > kT - 2) t0 = kT - 2;

  // Stage the 4 hd frames this WG can touch: frames t0..t0+3, 100 floats each.
  __shared__ float ahd[4 * kH];
  for (int i = threadIdx.x; i < 4 * kH; i += 256) {
    int m = i / kH;
    int h = i - m * kH;
    int t = t0 + m; if (t > kT - 1) t = kT - 1;
    ahd[i] = hd[(b * kH + h) * kT + t];
  }
  __syncthreads();

  const int  lane   = threadIdx.x & 31;
  const int  wave   = threadIdx.x >> 5;
  const int  n      = lane & 15;        // sample-in-tile (and C column)
  const bool hiHalf = lane >= 16;
  const int  s      = sBase + wave * 16 + n;

  // Linear interpolation indices (fp32, same as reference)
  float pos = (float)s * kR;
  int   i0  = (int)pos;
  if (i0 > kT - 2) i0 = kT - 2;
  float w = pos - (float)i0;
  int   m = i0 - t0;                    // row in {0,1,2}

  const float* f0b = f0 + b * kT;
  const float* psb = ps + b * kT;
  float f0v = f0b[i0];  f0v += (f0b[i0 + 1] - f0v) * w;
  float psv = psb[i0];  psv += (psb[i0 + 1] - psv) * w;

  // beta = 2*pi*f0*s/sr, range-reduced in double (f0*s/sr reaches ~2000;
  // fp32 reduction would cost ~1e-4*h radians of phase error at h=100)
  double u    = (double)f0v * ((double)s * (1.0 / 16000.0));
  float  beta = (float)((u - floor(u)) * 6.283185307179586);

  float twoC = 2.0f * __cosf(beta);
  float Sm1  = __sinf(psv);            // S_0 = sin(phi)
  float Scur = __sinf(beta + psv);     // S_1 = sin(beta + phi)

#if __has_builtin(__builtin_amdgcn_wmma_f32_16x16x4_f32)
  v8f c = {0.f, 0.f, 0.f, 0.f, 0.f, 0.f, 0.f, 0.f};
  const float rowMask = (n < 4) ? 1.0f : 0.0f;   // only A rows 0..3 are live
  const int   aRow    = (n & 3) * kH;            // clamped row (masked anyway)
  const int   colBase = hiHalf ? 2 : 0;          // K-slot layout per lane half
  for (int k = 0; k < 25; ++k) {
    // 4 recurrence steps -> harmonics 4k+1 .. 4k+4
    float v0 = Scur; float nx = __builtin_fmaf(twoC, Scur, -Sm1); Sm1 = Scur; Scur = nx;
    float v1 = Scur;       nx = __builtin_fmaf(twoC, Scur, -Sm1); Sm1 = Scur; Scur = nx;
    float v2 = Scur;       nx = __builtin_fmaf(twoC, Scur, -Sm1); Sm1 = Scur; Scur = nx;
    float v3 = Scur;       nx = __builtin_fmaf(twoC, Scur, -Sm1); Sm1 = Scur; Scur = nx;

    // B (4x16): VGPR0 = K0 (lanes 0-15) / K2 (lanes 16-31); VGPR1 = K1 / K3
    v2f bm; bm.x = hiHalf ? v2 : v0; bm.y = hiHalf ? v3 : v1;

    // A (16x4): lane = M row; VGPR0 = K0/K2, VGPR1 = K1/K3 (consecutive pair)
    const float* p = &ahd[aRow + 4 * k + colBase];
    v2f am; am.x = p[0] * rowMask; am.y = p[1] * rowMask;

    c = __builtin_amdgcn_wmma_f32_16x16x4_f32(
        /*neg_a=*/false, am, /*neg_b=*/false, bm,
        /*c_mod=*/(short)0, c, /*reuse_a=*/false, /*reuse_b=*/false);
  }
  if (!hiHalf) {
    // C layout: lanes 0-15, VGPR r holds C[M=r][N=lane]; rows m, m+1 are local
    float P0 = (m == 0) ? c[0] : (m == 1) ? c[1] : (m == 2) ? c[2] : c[3];
    float P1 = (m == 0) ? c[1] : (m == 1) ? c[2] : (m == 2) ? c[3] : c[4];
    out[b * kTs + s] = (1.0f - w) * P0 + w * P1;
  }
#else
  // Scalar fallback (still recurrence-based), lanes 0-15 only.
  if (!hiHalf) {
    float acc0 = 0.f, acc1 = 0.f;
    const float* r0 = &ahd[m * kH];
    const float* r1 = &ahd[(m + 1) * kH];
    for (int h = 0; h < kH; ++h) {
      float S  = Scur;
      float nx = __builtin_fmaf(twoC, Scur, -Sm1); Sm1 = Scur; Scur = nx;
      acc0 = __builtin_fmaf(S, r0[h], acc0);
      acc1 = __builtin_fmaf(S, r1[h], acc1);
    }
    out[b * kTs + s] = (1.0f - w) * acc0 + w * acc1;
  }
#endif
}

extern "C" void kernel_launch(void* const* d_in, const int* in_sizes, int n_in,
                              void* d_out, int out_size, void* d_ws, size_t ws_size,
                              hipStream_t stream) {
  (void)in_sizes; (void)n_in; (void)out_size; (void)d_ws; (void)ws_size;
  const float* f0 = (const float*)d_in[0];   // [8,1,1000]
  const float* hd = (const float*)d_in[1];   // [8,100,1000]
  const float* ps = (const float*)d_in[2];   // [8,1,1000]
  float* out = (float*)d_out;                // [8,80000]
  dim3 grid(kB * kChunksPerB);               // 5000 blocks
  harmosc_kernel<<<grid, 256, 0, stream>>>(f0, hd, ps, out);
}